// SelectiveSSM_65687229825599
// MI455X (gfx1250) — compile-verified
//
#include <hip/hip_runtime.h>

#define D_INNER   2048
#define N_STATE   16
#define DT_RANK   64
#define BATCH     2
#define SEQLEN    2048
#define M_ROWS    (BATCH * SEQLEN)          // 4096
#define PROJ_DIM  (DT_RANK + 2 * N_STATE)   // 96
#define CHUNK     128
#define NCHUNK    (SEQLEN / CHUNK)          // 16

typedef float v2f __attribute__((ext_vector_type(2)));
typedef float v8f __attribute__((ext_vector_type(8)));
typedef unsigned int u32x4 __attribute__((ext_vector_type(4)));
typedef int i32x4 __attribute__((ext_vector_type(4)));
typedef int i32x8 __attribute__((ext_vector_type(8)));

// -------------------------------------------------------------------------
// TDM: stage proj[b, l0:l0+CHUNK, 64:96] (2D strided tile, 128 x 32 f32)
// into LDS with one Tensor Data Mover op.  D# packing per CDNA5 ISA ch.8.
// Caller: exactly one lane of one wave per workgroup; follow with barrier.
// -------------------------------------------------------------------------
__device__ __forceinline__ void tdm_stage_bc(const float* proj, int b, int l0,
                                             float* lds_dst) {
  const unsigned long long ga = (unsigned long long)(uintptr_t)(
      proj + ((size_t)b * SEQLEN + l0) * PROJ_DIM + DT_RANK);
  const unsigned int lds_addr = (unsigned int)(uintptr_t)lds_dst;  // LDS = flat[31:0]

  u32x4 g0;
  g0[0] = 1u;                                                    // count=1 (valid user D#)
  g0[1] = lds_addr;                                              // lds_addr
  g0[2] = (unsigned int)ga;                                      // global_addr[31:0]
  g0[3] = ((unsigned int)(ga >> 32) & 0x01FFFFFFu) | (2u << 30); // addr[56:32] | type=2

  i32x8 g1;
  g1[0] = (int)(2u << 16);            // workgroup_mask=0, data_size=2 (4 bytes)
  g1[1] = (int)(32u << 16);           // tensor_dim0 = 32 (bits 79:48, low half)
  g1[2] = (int)((unsigned)CHUNK << 16); // tensor_dim1 = 128 (bits 111:80, low half)
  g1[3] = (int)(32u << 16);           // tile_dim0 = 32 (bits 127:112)
  g1[4] = (int)CHUNK;                 // tile_dim1 = 128 (bits 143:128), tile_dim2 = 0
  g1[5] = (int)PROJ_DIM;              // tensor_dim0_stride = 96 (bits 207:160, low 32)
  g1[6] = 0;                          // stride0 hi, tensor_dim1_stride lo (unused, 2D)
  g1[7] = 0;

  i32x4 z4 = {0, 0, 0, 0};            // groups 2/3: tile_dim3/4 = 0 -> unused (2D)
#if defined(__clang_major__) && (__clang_major__ >= 23)
  i32x8 z8 = {0, 0, 0, 0, 0, 0, 0, 0};
  __builtin_amdgcn_tensor_load_to_lds(g0, g1, z4, z4, z8, 0);
#else
  __builtin_amdgcn_tensor_load_to_lds(g0, g1, z4, z4, 0);
#endif
  __builtin_amdgcn_s_wait_tensorcnt(0);
}

// -------------------------------------------------------------------------
// Kernel 1: projected[M,96] = x[M,2048] @ W_x^T        (f32 WMMA 16x16x4)
// One wave computes one 16x16 output tile.
// -------------------------------------------------------------------------
__global__ void __launch_bounds__(32)
proj_wmma_kernel(const float* __restrict__ x, const float* __restrict__ Wx,
                 float* __restrict__ proj) {
  const int mt   = blockIdx.x;            // 256 row tiles
  const int nt   = blockIdx.y;            // 6 col tiles
  const int lane = threadIdx.x;
  const int lm   = lane & 15;
  const int half = lane >> 4;

  const float* arow = x  + (size_t)(mt * 16 + lm) * D_INNER;   // x row (M)
  const float* brow = Wx + (size_t)(nt * 16 + lm) * D_INNER;   // W_x row (=col of B)

  v8f acc = {};
  for (int k = 0; k < D_INNER; k += 4) {
    v2f a = *(const v2f*)(arow + k + 2 * half);   // K = 2*half, 2*half+1
    v2f b = *(const v2f*)(brow + k + 2 * half);
    acc = __builtin_amdgcn_wmma_f32_16x16x4_f32(
        /*neg_a=*/false, a, /*neg_b=*/false, b,
        /*c_mod=*/(short)0, acc, /*reuse_a=*/false, /*reuse_b=*/false);
  }
#pragma unroll
  for (int r = 0; r < 8; ++r) {
    int m = mt * 16 + r + 8 * half;               // C layout: VGPR r -> rows r, r+8
    proj[(size_t)m * PROJ_DIM + nt * 16 + lm] = acc[r];
  }
}

// -------------------------------------------------------------------------
// Kernel 2: delta[M,2048] = softplus(delta_raw[M,64] @ W_dt^T + b_dt)
// -------------------------------------------------------------------------
__global__ void __launch_bounds__(32)
delta_wmma_kernel(const float* __restrict__ proj, const float* __restrict__ Wdt,
                  const float* __restrict__ bdt, float* __restrict__ delta) {
  const int mt   = blockIdx.x;            // 256 row tiles
  const int nt   = blockIdx.y;            // 128 col tiles
  const int lane = threadIdx.x;
  const int lm   = lane & 15;
  const int half = lane >> 4;

  const float* arow = proj + (size_t)(mt * 16 + lm) * PROJ_DIM;   // delta_raw row
  const float* brow = Wdt  + (size_t)(nt * 16 + lm) * DT_RANK;    // W_dt row

  v8f acc = {};
#pragma unroll
  for (int k = 0; k < DT_RANK; k += 4) {
    v2f a = *(const v2f*)(arow + k + 2 * half);
    v2f b = *(const v2f*)(brow + k + 2 * half);
    acc = __builtin_amdgcn_wmma_f32_16x16x4_f32(
        false, a, false, b, (short)0, acc, false, false);
  }

  const float bias = bdt[nt * 16 + lm];
#pragma unroll
  for (int r = 0; r < 8; ++r) {
    int m = mt * 16 + r + 8 * half;
    float v = acc[r] + bias;
    float sp = (v > 20.0f) ? v : __logf(1.0f + __expf(v));   // softplus
    delta[(size_t)m * D_INNER + nt * 16 + lm] = sp;
  }
}

// -------------------------------------------------------------------------
// Kernel 3: chunk-local scan (h starts at 0).  One thread per (b, d, chunk).
// B rows staged into LDS by the TDM; inner loop reads are broadcast DS loads.
// -------------------------------------------------------------------------
__global__ void __launch_bounds__(256)
scan_pass1_kernel(const float* __restrict__ x, const float* __restrict__ delta,
                  const float* __restrict__ proj, const float* __restrict__ A_log,
                  float* __restrict__ S, float* __restrict__ sumdt_ws) {
  __shared__ float bc[CHUNK * 32];                // B|C rows for this chunk (16 KB)
  const int d = blockIdx.x * 256 + threadIdx.x;   // 0..2047
  const int c = blockIdx.y;                       // chunk
  const int b = blockIdx.z;                       // batch
  const int l0 = c * CHUNK;

  if (threadIdx.x == 0) tdm_stage_bc(proj, b, l0, bc);   // TDM + s_wait_tensorcnt
  __syncthreads();

  float Arow[N_STATE];
#pragma unroll
  for (int n = 0; n < N_STATE; ++n)
    Arow[n] = -__expf(A_log[(size_t)d * N_STATE + n]);

  float h[N_STATE];
#pragma unroll
  for (int n = 0; n < N_STATE; ++n) h[n] = 0.0f;
  float sumdt = 0.0f;

  for (int l = l0; l < l0 + CHUNK; ++l) {
    const size_t row = (size_t)b * SEQLEN + l;
    const float* dptr = delta + row * D_INNER + d;
    const float* uptr = x     + row * D_INNER + d;
    __builtin_prefetch(dptr + D_INNER, 0, 0);     // global_prefetch_b8
    __builtin_prefetch(uptr + D_INNER, 0, 0);
    const float dt  = *dptr;
    const float u   = *uptr;
    const float dtu = dt * u;
    const float* pr = bc + (size_t)(l - l0) * 32; // LDS broadcast reads
    sumdt += dt;
#pragma unroll
    for (int n = 0; n < N_STATE; ++n) {
      float dA = __expf(dt * Arow[n]);
      h[n] = fmaf(dA, h[n], dtu * pr[n]);
    }
  }

  const size_t sbase = (((size_t)b * D_INNER + d) * NCHUNK + c) * N_STATE;
#pragma unroll
  for (int n = 0; n < N_STATE; ++n) S[sbase + n] = h[n];
  sumdt_ws[((size_t)b * D_INNER + d) * NCHUNK + c] = sumdt;
}

// -------------------------------------------------------------------------
// Kernel 4: sequential combine across the 16 chunk summaries per channel.
// -------------------------------------------------------------------------
__global__ void __launch_bounds__(256)
scan_combine_kernel(const float* __restrict__ A_log, const float* __restrict__ S,
                    const float* __restrict__ sumdt_ws, float* __restrict__ Hin) {
  const int idx = blockIdx.x * 256 + threadIdx.x;   // b*D_INNER + d
  const int d = idx & (D_INNER - 1);

  float Arow[N_STATE];
#pragma unroll
  for (int n = 0; n < N_STATE; ++n)
    Arow[n] = -__expf(A_log[(size_t)d * N_STATE + n]);

  float h[N_STATE];
#pragma unroll
  for (int n = 0; n < N_STATE; ++n) h[n] = 0.0f;

  for (int c = 0; c < NCHUNK; ++c) {
    const size_t base = ((size_t)idx * NCHUNK + c) * N_STATE;
#pragma unroll
    for (int n = 0; n < N_STATE; ++n) Hin[base + n] = h[n];
    const float sd = sumdt_ws[(size_t)idx * NCHUNK + c];
#pragma unroll
    for (int n = 0; n < N_STATE; ++n) {
      float P = __expf(sd * Arow[n]);
      h[n] = fmaf(P, h[n], S[base + n]);
    }
  }
}

// -------------------------------------------------------------------------
// Kernel 5: final scan seeded with Hin; emits y + x*D_skip.
// -------------------------------------------------------------------------
__global__ void __launch_bounds__(256)
scan_pass2_kernel(const float* __restrict__ x, const float* __restrict__ delta,
                  const float* __restrict__ proj, const float* __restrict__ A_log,
                  const float* __restrict__ Dskip, const float* __restrict__ Hin,
                  float* __restrict__ out) {
  __shared__ float bc[CHUNK * 32];                // B|C rows for this chunk (16 KB)
  const int d = blockIdx.x * 256 + threadIdx.x;
  const int c = blockIdx.y;
  const int b = blockIdx.z;
  const int l0 = c * CHUNK;

  if (threadIdx.x == 0) tdm_stage_bc(proj, b, l0, bc);   // TDM + s_wait_tensorcnt
  __syncthreads();

  float Arow[N_STATE];
#pragma unroll
  for (int n = 0; n < N_STATE; ++n)
    Arow[n] = -__expf(A_log[(size_t)d * N_STATE + n]);

  float h[N_STATE];
  const size_t hbase = (((size_t)b * D_INNER + d) * NCHUNK + c) * N_STATE;
#pragma unroll
  for (int n = 0; n < N_STATE; ++n) h[n] = Hin[hbase + n];

  const float dskip = Dskip[d];
  for (int l = l0; l < l0 + CHUNK; ++l) {
    const size_t row = (size_t)b * SEQLEN + l;
    const float* dptr = delta + row * D_INNER + d;
    const float* uptr = x     + row * D_INNER + d;
    __builtin_prefetch(dptr + D_INNER, 0, 0);
    __builtin_prefetch(uptr + D_INNER, 0, 0);
    const float dt  = *dptr;
    const float u   = *uptr;
    const float dtu = dt * u;
    const float* pr = bc + (size_t)(l - l0) * 32; // LDS: B at [0:16), C at [16:32)
    float y = 0.0f;
#pragma unroll
    for (int n = 0; n < N_STATE; ++n) {
      float dA = __expf(dt * Arow[n]);
      h[n] = fmaf(dA, h[n], dtu * pr[n]);
      y = fmaf(h[n], pr[16 + n], y);
    }
    out[row * D_INNER + d] = fmaf(u, dskip, y);
  }
}

// -------------------------------------------------------------------------
extern "C" void kernel_launch(void* const* d_in, const int* in_sizes, int n_in,
                              void* d_out, int out_size, void* d_ws, size_t ws_size,
                              hipStream_t stream) {
  const float* x      = (const float*)d_in[0];
  const float* A_log  = (const float*)d_in[1];
  const float* D_skip = (const float*)d_in[2];
  const float* W_x    = (const float*)d_in[3];
  const float* W_dt   = (const float*)d_in[4];
  const float* b_dt   = (const float*)d_in[5];
  float* out = (float*)d_out;

  // Workspace carve-up (floats):
  float* ws       = (float*)d_ws;
  float* proj     = ws;                                           // 4096*96
  float* delta    = proj  + (size_t)M_ROWS * PROJ_DIM;            // 4096*2048
  float* S        = delta + (size_t)M_ROWS * D_INNER;             // 2*2048*16*16
  float* sumdt_ws = S     + (size_t)BATCH * D_INNER * NCHUNK * N_STATE; // 2*2048*16
  float* Hin      = sumdt_ws + (size_t)BATCH * D_INNER * NCHUNK;  // 2*2048*16*16

  // 1) projection GEMM (WMMA f32)
  proj_wmma_kernel<<<dim3(M_ROWS / 16, PROJ_DIM / 16), 32, 0, stream>>>(x, W_x, proj);

  // 2) delta GEMM + softplus (WMMA f32)
  delta_wmma_kernel<<<dim3(M_ROWS / 16, D_INNER / 16), 32, 0, stream>>>(proj, W_dt, b_dt, delta);

  // 3) chunk-local scans (TDM-staged B rows)
  scan_pass1_kernel<<<dim3(D_INNER / 256, NCHUNK, BATCH), 256, 0, stream>>>(
      x, delta, proj, A_log, S, sumdt_ws);

  // 4) inter-chunk combine
  scan_combine_kernel<<<dim3((BATCH * D_INNER) / 256), 256, 0, stream>>>(
      A_log, S, sumdt_ws, Hin);

  // 5) seeded scan + epilogue (TDM-staged B/C rows)
  scan_pass2_kernel<<<dim3(D_INNER / 256, NCHUNK, BATCH), 256, 0, stream>>>(
      x, delta, proj, A_log, D_skip, Hin, out);
}